// GraphAttentionLayer_22720376996171
// MI455X (gfx1250) — compile-verified
//
#include <hip/hip_runtime.h>
#include <math.h>

#define B_   32
#define P_   128
#define L_   19
#define H_   256
#define HA_  128
#define SLOPE 0.2f
#define NEGV  (-1000000000.0f)

#define WSTRIDE 129   // 128 + 1 pad: column reads hit distinct banks
#define ASTRIDE 264   // 256 + 8 pad: half-wave B-frag reads hit distinct banks

typedef __attribute__((ext_vector_type(2))) float v2f;
typedef __attribute__((ext_vector_type(8))) float v8f;

// ---------------- Kernel 0: v_a = W1^T w2a, v_b = W1^T w2b (tiny) -----------
__global__ void gat_fold_w2(const float* __restrict__ W1,
                            const float* __restrict__ W2,
                            float* __restrict__ va, float* __restrict__ vb) {
    int h = threadIdx.x;             // 256 threads, one per h
    float sa = 0.f, sb = 0.f;
    #pragma unroll 8
    for (int a = 0; a < HA_; ++a) {
        float w = W1[a * H_ + h];    // coalesced across h
        sa = fmaf(w, W2[a], sa);
        sb = fmaf(w, W2[HA_ + a], sb);
    }
    va[h] = sa;
    vb[h] = sb;
}

// ---------------- Kernel 1: fully fused e -> softmax -> WMMA GEMM -----------
// one block per (b,l); 256 threads = 8 waves; wave w owns output rows 16w..16w+15
__global__ void __launch_bounds__(256, 1)
gat_main(const float* __restrict__ agents,
         const float* __restrict__ adj,
         const unsigned char* __restrict__ mask,
         const float* __restrict__ va,
         const float* __restrict__ vb,
         float* __restrict__ out) {
    extern __shared__ float smem[];
    float* Wlds = smem;                          // 128 x WSTRIDE
    float* Ag   = Wlds + P_ * WSTRIDE;           // 128 x ASTRIDE
    float* eil  = Ag + P_ * ASTRIDE;             // 128
    float* ejl  = eil + P_;                      // 128
    float* mfl  = ejl + P_;                      // 128 (0/1)

    const int bl = blockIdx.x;
    const int b  = bl / L_;
    const int l  = bl - b * L_;
    const int tid = threadIdx.x;

    // Stage agents[b,:,l,:] -> LDS (row-coalesced float4); agents read ONCE from HBM
    for (int idx = tid; idx < P_ * (H_ / 4); idx += 256) {
        int j  = idx >> 6;                       // /64 float4 per row
        int c4 = (idx & 63) << 2;
        float4 v = *(const float4*)(agents + ((size_t)(b * P_ + j) * L_ + l) * H_ + c4);
        float* dst = Ag + j * ASTRIDE + c4;
        dst[0] = v.x; dst[1] = v.y; dst[2] = v.z; dst[3] = v.w;
    }
    if (tid < P_) {
        mfl[tid] = mask[b * P_ + tid] ? 1.0f : 0.0f;
    }
    __syncthreads();

    const int w    = tid >> 5;
    const int lane = tid & 31;

    // ---- e_i/e_j from the staged panel: e[i] = Ag[i,:]·v_{a,b} --------------
    {
        const int base = lane * 8;               // 32 lanes x 8 = 256
        float4 va0 = *(const float4*)(va + base);
        float4 va1 = *(const float4*)(va + base + 4);
        float4 vb0 = *(const float4*)(vb + base);
        float4 vb1 = *(const float4*)(vb + base + 4);
        for (int rr = 0; rr < 16; ++rr) {
            const int i = w * 16 + rr;
            const float* row = Ag + i * ASTRIDE + base;
            float4 a0 = *(const float4*)(row);
            float4 a1 = *(const float4*)(row + 4);
            float si = a0.x*va0.x + a0.y*va0.y + a0.z*va0.z + a0.w*va0.w
                     + a1.x*va1.x + a1.y*va1.y + a1.z*va1.z + a1.w*va1.w;
            float sj = a0.x*vb0.x + a0.y*vb0.y + a0.z*vb0.z + a0.w*vb0.w
                     + a1.x*vb1.x + a1.y*vb1.y + a1.z*vb1.z + a1.w*vb1.w;
            #pragma unroll
            for (int off = 16; off > 0; off >>= 1) {
                si += __shfl_xor(si, off, 32);
                sj += __shfl_xor(sj, off, 32);
            }
            if (lane == 0) { eil[i] = si; ejl[i] = sj; }
        }
    }
    __syncthreads();   // softmax row i needs ejl[j] for all j

    // ---- softmax over j for rows i = 16w .. 16w+15, fused with adj gating ----
    for (int rr = 0; rr < 16; ++rr) {
        const int i = w * 16 + rr;
        const float evi = eil[i];
        float lg[4];
        #pragma unroll
        for (int q = 0; q < 4; ++q) {
            int j = lane + q * 32;
            float x = evi + ejl[j];
            x = (x > 0.f) ? x : SLOPE * x;       // leaky relu
            lg[q] = (mfl[j] != 0.f) ? NEGV : x;  // mask over j
        }
        float m = fmaxf(fmaxf(lg[0], lg[1]), fmaxf(lg[2], lg[3]));
        #pragma unroll
        for (int off = 16; off > 0; off >>= 1) m = fmaxf(m, __shfl_xor(m, off, 32));
        float p[4], s = 0.f;
        #pragma unroll
        for (int q = 0; q < 4; ++q) { p[q] = __expf(lg[q] - m); s += p[q]; }
        #pragma unroll
        for (int off = 16; off > 0; off >>= 1) s += __shfl_xor(s, off, 32);
        const float inv = 1.0f / s;
        #pragma unroll
        for (int q = 0; q < 4; ++q) {
            int j = lane + q * 32;
            float av = adj[((size_t)(b * P_ + i) * P_ + j) * L_ + l];
            Wlds[i * WSTRIDE + j] = (av > 0.f) ? p[q] * inv : 0.f;
        }
    }
    // No barrier: wave w's GEMM consumes only the Wlds rows it just wrote,
    // plus Ag which was fenced by the first __syncthreads().

    // ---- WMMA f32 GEMM: out[16w.., :] = Wlds[16w..,:] x Ag, fused sigmoid ----
    const int nlo  = lane & 15;
    const int koff = (lane >> 4) << 1;           // lanes 0-15: K=0,1; 16-31: K=2,3

    // Preload the wave's 16x128 A panel as 32 fragments (ISA 16x4 f32 A layout)
    const float* arow = Wlds + (w * 16 + nlo) * WSTRIDE + koff;
    v2f afr[32];
    #pragma unroll
    for (int kk = 0; kk < 32; ++kk) {
        afr[kk].x = arow[kk * 4];
        afr[kk].y = arow[kk * 4 + 1];
    }

    for (int th = 0; th < 16; ++th) {            // 16 column tiles of 16
        v8f acc = {};
        const float* bcol = Ag + koff * ASTRIDE + th * 16 + nlo;
        #pragma unroll
        for (int kk = 0; kk < 32; ++kk) {
            v2f bfr;
            bfr.x = bcol[(kk * 4) * ASTRIDE];
            bfr.y = bcol[(kk * 4 + 1) * ASTRIDE];
            acc = __builtin_amdgcn_wmma_f32_16x16x4_f32(
                false, afr[kk], false, bfr, (short)0, acc, false, false);
        }
        // D layout: VGPR r -> M=r (lanes 0-15), M=r+8 (lanes 16-31); N = lane&15
        const int hcol = th * 16 + nlo;
        const int ibase = w * 16 + ((lane >> 4) << 3);
        #pragma unroll
        for (int r = 0; r < 8; ++r) {
            float x = acc[r];
            float sg = 1.0f / (1.0f + __expf(-x));
            out[((size_t)(b * P_ + ibase + r) * L_ + l) * H_ + hcol] = sg;
        }
    }
}

// ---------------------------------------------------------------------------
extern "C" void kernel_launch(void* const* d_in, const int* in_sizes, int n_in,
                              void* d_out, int out_size, void* d_ws, size_t ws_size,
                              hipStream_t stream) {
    const float* agents        = (const float*)d_in[0];
    const float* adj           = (const float*)d_in[1];
    const unsigned char* mask  = (const unsigned char*)d_in[2]; // jnp bool -> 1B
    const float* W1            = (const float*)d_in[3];
    const float* W2            = (const float*)d_in[4];
    float* out = (float*)d_out;

    float* ws = (float*)d_ws;
    float* va = ws;                              // 256
    float* vb = ws + 256;                        // 256

    gat_fold_w2<<<1, 256, 0, stream>>>(W1, W2, va, vb);

    const size_t shbytes = (size_t)(P_ * WSTRIDE + P_ * ASTRIDE + 3 * P_) * sizeof(float);
    gat_main<<<B_ * L_, 256, shbytes, stream>>>(agents, adj, mask, va, vb, out);
}